// CCModel_35742717837614
// MI455X (gfx1250) — compile-verified
//
#include <hip/hip_runtime.h>

typedef float v2f __attribute__((ext_vector_type(2)));
typedef float v8f __attribute__((ext_vector_type(8)));

#define NT    3000   // samples per channel
#define NOUT  603    // 2*NLAG + 3 output lags
#define XOFF  301    // left zero-pad of data1 (lag offset)
#define XPAD  3775   // covers x index range [-301, 3473]
#define YOFF  240    // left zero-pad of data2 (15 blocks * 16)
#define YPAD  3488   // covers y index range [-240, 3247]
#define NS    203    // s-steps per tile: 16 (column shifts) + 187 (time blocks)

// One block per channel (2048 blocks), 96 threads = 3 wave32.
// Each wave computes one 16x16 f32 WMMA accumulator tile = 256 consecutive lags:
//   C[oi][n] = sum_s A_s(16x16 Hankel of x) * B_s(16 shifted segments of y)
//            = out[16*(OB0+n) + oi]
__global__ __launch_bounds__(96) void xcorr_wmma_f32(const float* __restrict__ d1,
                                                     const float* __restrict__ d2,
                                                     float* __restrict__ out) {
  __shared__ float xs[XPAD];
  __shared__ float ys[YPAD];

  const int c   = blockIdx.x;
  const int tid = threadIdx.x;
  const float* x = d1 + c * NT;
  const float* y = d2 + c * NT;

  // Cooperative zero-padded staging of both signals into LDS.
  for (int i = tid; i < XPAD; i += 96) {
    int t = i - XOFF;
    xs[i] = (t >= 0 && t < NT) ? x[t] : 0.0f;
  }
  for (int i = tid; i < YPAD; i += 96) {
    int t = i - YOFF;
    ys[i] = (t >= 0 && t < NT) ? y[t] : 0.0f;
  }
  __syncthreads();

  const int wave = tid >> 5;         // 0..2 : which 256-lag tile
  const int lane = tid & 31;
  const int m    = lane & 15;        // A row (oi) / B column (n)
  const int kb2  = (lane >> 4) << 1; // K base: lanes 0-15 -> K{0,1}, 16-31 -> K{2,3}
  const int OB0  = wave << 4;        // first lag-block index of this tile

  // Per-lane constant address components.
  const int alo = m + kb2;                        // A: x[16*s + oi + k - 301]
  const int ylo = YOFF - ((OB0 + m) << 4) + kb2;  // B: y[16*(s-OB0-n) + k]

  v8f acc = {};
  int base = OB0 << 4;               // 16 * s_abs
  for (int s = 0; s < NS; ++s) {
    const int xb = base + alo;
    const int yb = base + ylo;
#pragma unroll
    for (int kc = 0; kc < 4; ++kc) { // four K=4 slices cover ti = 0..15
      v2f a, b;
      a.x = xs[xb + 4 * kc];
      a.y = xs[xb + 4 * kc + 1];
      b.x = ys[yb + 4 * kc];
      b.y = ys[yb + 4 * kc + 1];
      // D = A(16x4 f32) * B(4x16 f32) + C, full fp32 precision
      acc = __builtin_amdgcn_wmma_f32_16x16x4_f32(false, a, false, b,
                                                  (short)0, acc, false, false);
    }
    base += 16;
  }

  // C/D layout: lane l, VGPR i -> C[i + 8*(l>>4)][l&15]
  const int hi = lane >> 4;
#pragma unroll
  for (int i = 0; i < 8; ++i) {
    int oi = i + (hi << 3);
    int o  = ((OB0 + m) << 4) + oi;  // lag index = 16*(OB0+n) + oi
    if (o < NOUT) out[c * NOUT + o] = acc[i];
  }
}

extern "C" void kernel_launch(void* const* d_in, const int* in_sizes, int n_in,
                              void* d_out, int out_size, void* d_ws, size_t ws_size,
                              hipStream_t stream) {
  const float* d1 = (const float*)d_in[0];  // (64,32,3000) f32
  const float* d2 = (const float*)d_in[1];  // (64,32,3000) f32
  float* out = (float*)d_out;               // (64,32,603)  f32
  (void)in_sizes; (void)n_in; (void)out_size; (void)d_ws; (void)ws_size;
  hipLaunchKernelGGL(xcorr_wmma_f32, dim3(2048), dim3(96), 0, stream, d1, d2, out);
}